// RNNTLoss_266287972582
// MI455X (gfx1250) — compile-verified
//
#include <hip/hip_runtime.h>
#include <hip/hip_bf16.h>

typedef __attribute__((ext_vector_type(4))) unsigned int v4u;
typedef __attribute__((ext_vector_type(8))) int          v8i;
typedef __attribute__((ext_vector_type(4))) int          v4i;

#define RNNT_B 8
#define RNNT_T 256
#define RNNT_U 65
#define RNNT_V 1024
#define ROWS_PER_BLK 16   // 133120 / 16 = 8320 blocks

__device__ __forceinline__ float wave_rmax(float v) {
#pragma unroll
  for (int m = 16; m > 0; m >>= 1) v = fmaxf(v, __shfl_xor(v, m, 32));
  return v;
}
__device__ __forceinline__ float wave_rsum(float v) {
#pragma unroll
  for (int m = 16; m > 0; m >>= 1) v += __shfl_xor(v, m, 32);
  return v;
}

// Issue one TDM DMA: 1024-f32 row (global -> LDS buffer). Called by wave 0 only.
__device__ __forceinline__ void tdm_issue_row(const float* rowp, unsigned lds_off) {
  unsigned long long ga = (unsigned long long)(size_t)rowp;
  v4u g0 = { 1u,                                          // count=1, user descriptor
             lds_off,                                     // lds_addr (bytes)
             (unsigned)(ga & 0xFFFFFFFFull),              // global_addr[31:0]
             (unsigned)((ga >> 32) & 0x01FFFFFFull)       // global_addr[56:32]
               | 0x80000000u };                           // type=2 ("image")
  v8i g1 = { (int)0x00020000,                             // data_size=2 (4 bytes), no multicast
             (int)(RNNT_V << 16),                         // tensor_dim0[15:0] in [63:48]
             (int)(1u << 16),                             // tensor_dim0 hi=0; tensor_dim1=1
             (int)(RNNT_V << 16),                         // tensor_dim1 hi=0; tile_dim0=1024
             0,                                           // tile_dim1=0 (1-D tile), tile_dim2=0
             RNNT_V,                                      // tensor_dim0_stride[31:0]
             0, 0 };
  v4i gz4 = {0, 0, 0, 0};
  v8i gz8 = {0, 0, 0, 0, 0, 0, 0, 0};
  __builtin_amdgcn_tensor_load_to_lds(g0, g1, gz4, gz4, gz8, 0);
}

// Phase 1: each block handles 16 consecutive (b,t,u) rows with a 2-deep
// TDM pipeline: DMA of row i+1 overlaps the log-sum-exp of row i.
__global__ __launch_bounds__(256) void rnnt_lse_kernel(
    const float* __restrict__ acts, const int* __restrict__ labels,
    float* __restrict__ lpb, float* __restrict__ lpl) {
  __shared__ float sh[2][RNNT_V];
  __shared__ float redm[8];
  __shared__ float reds[8];

  const int  base_row = blockIdx.x * ROWS_PER_BLK;
  const int  wid  = threadIdx.x >> 5;
  const int  lane = threadIdx.x & 31;
  const bool w0   = (threadIdx.x < 32);

  // prime the pipeline: row 0 -> buffer 0
  if (w0) {
    tdm_issue_row(acts + (size_t)base_row * RNNT_V,
                  (unsigned)(size_t)(&sh[0][0]));
  }

#pragma unroll 1
  for (int i = 0; i < ROWS_PER_BLK; ++i) {
    const int row = base_row + i;
    const int buf = i & 1;

    if (w0) {
      if (i + 1 < ROWS_PER_BLK) {
        tdm_issue_row(acts + (size_t)(row + 1) * RNNT_V,
                      (unsigned)(size_t)(&sh[buf ^ 1][0]));
        __builtin_amdgcn_s_wait_tensorcnt(1);   // oldest (row i) complete; row i+1 in flight
      } else {
        __builtin_amdgcn_s_wait_tensorcnt(0);
      }
    }
    __syncthreads();                            // row i now visible to all waves

    // ---- block log-sum-exp over 1024 values (4 per thread) ----
    const float4 x = ((const float4*)sh[buf])[threadIdx.x];

    float wm = wave_rmax(fmaxf(fmaxf(x.x, x.y), fmaxf(x.z, x.w)));
    if (lane == 0) redm[wid] = wm;
    __syncthreads();
    float bm = redm[0];
#pragma unroll
    for (int k = 1; k < 8; ++k) bm = fmaxf(bm, redm[k]);

    float s = expf(x.x - bm) + expf(x.y - bm) + expf(x.z - bm) + expf(x.w - bm);
    float ws = wave_rsum(s);
    if (lane == 0) reds[wid] = ws;
    __syncthreads();
    float bs = reds[0];
#pragma unroll
    for (int k = 1; k < 8; ++k) bs += reds[k];

    const float lse = bm + logf(bs);

    const int b = row / (RNNT_T * RNNT_U);
    const int u = row % RNNT_U;
    if (threadIdx.x == 0) lpb[row] = x.x - lse;           // BLANK = index 0
    if (u < RNNT_U - 1) {
      const int lab = labels[b * (RNNT_U - 1) + u];       // in [1, V)
      if ((lab >> 2) == (int)threadIdx.x) {
        const int c = lab & 3;
        const float xv = (c == 0) ? x.x : (c == 1) ? x.y : (c == 2) ? x.z : x.w;
        lpl[row] = xv - lse;                              // FastEmit is fwd no-op
      }
    }
    __syncthreads();   // buffer 'buf' free before it is re-targeted at i+1's issue
  }
}

__device__ __forceinline__ float laexp(float a, float b) {
  const float m = fmaxf(a, b);
  return m + log1pf(expf(-fabsf(a - b)));
}

// Phase 2: anti-diagonal wavefront DP, one block per batch element.
// alpha[t][u] = logaddexp(alpha[t-1][u] + lpb[t-1][u], alpha[t][u-1] + lpl[t][u-1])
__global__ __launch_bounds__(96) void rnnt_dp_kernel(
    const float* __restrict__ lpb, const float* __restrict__ lpl,
    const int* __restrict__ act_lens, const int* __restrict__ label_lens,
    float* __restrict__ costs) {
  const int b = blockIdx.x;
  const int u = threadIdx.x;
  __shared__ float bufA[RNNT_U];
  __shared__ float bufB[RNNT_U];
  __shared__ float result;

  const int tend = act_lens[b] - 1;
  const int uend = label_lens[b];

  float* cur  = bufA;
  float* prev = bufB;

  for (int d = 0; d < RNNT_T + RNNT_U - 1; ++d) {
    float val = 0.0f;
    const bool active = (u < RNNT_U) && (u <= d) && (d - u < RNNT_T);
    if (active) {
      const int t = d - u;
      if (d == 0) {
        val = 0.0f;
      } else if (t == 0) {
        val = prev[u - 1] + lpl[(b * RNNT_T + 0) * RNNT_U + (u - 1)];
      } else if (u == 0) {
        val = prev[0] + lpb[(b * RNNT_T + (t - 1)) * RNNT_U + 0];
      } else {
        val = laexp(prev[u]     + lpb[(b * RNNT_T + (t - 1)) * RNNT_U + u],
                    prev[u - 1] + lpl[(b * RNNT_T + t)       * RNNT_U + (u - 1)]);
      }
      cur[u] = val;
      if (d == tend + uend && u == uend)
        result = -(val + lpb[(b * RNNT_T + tend) * RNNT_U + uend]);
    }
    __syncthreads();
    float* tmp = prev; prev = cur; cur = tmp;
  }
  if (threadIdx.x == 0) costs[b] = result;
}

__global__ void rnnt_final_kernel(const float* __restrict__ costs, float* __restrict__ out) {
  if (threadIdx.x == 0) {
    float s = 0.0f;
#pragma unroll
    for (int i = 0; i < RNNT_B; ++i) s += costs[i];
    out[0] = s / (float)RNNT_B;
  }
}

extern "C" void kernel_launch(void* const* d_in, const int* in_sizes, int n_in,
                              void* d_out, int out_size, void* d_ws, size_t ws_size,
                              hipStream_t stream) {
  const float* acts      = (const float*)d_in[0];
  const int*   labels    = (const int*)d_in[1];
  const int*   act_lens  = (const int*)d_in[2];
  const int*   label_lens= (const int*)d_in[3];

  const size_t nrows = (size_t)RNNT_B * RNNT_T * RNNT_U;   // 133120
  float* lpb   = (float*)d_ws;
  float* lpl   = lpb + nrows;
  float* costs = lpl + nrows;

  rnnt_lse_kernel<<<(unsigned)(nrows / ROWS_PER_BLK), 256, 0, stream>>>(acts, labels, lpb, lpl);
  rnnt_dp_kernel<<<RNNT_B, 96, 0, stream>>>(lpb, lpl, act_lens, label_lens, costs);
  rnnt_final_kernel<<<1, 32, 0, stream>>>(costs, (float*)d_out);
}